// R_Actor_43765716747017
// MI455X (gfx1250) — compile-verified
//
#include <hip/hip_runtime.h>
#include <hip/hip_bf16.h>

// Problem constants (from reference)
#define N_ROLL   256
#define N_AGENT  64
#define HIDDEN   256
#define CL_DIST  0.5f

// Tiling
#define AG_PER_BLK 16
#define SPLIT      (N_AGENT / AG_PER_BLK)           // 4 -> 1024 blocks
#define HP         272                               // padded LDS row stride (floats); 272%64==16 -> conflict-free half-wave reads
#define SMEM_BYTES ((N_AGENT * HP + AG_PER_BLK * N_AGENT) * 4)  // 73728 B (<320KB WGP LDS)

typedef __attribute__((ext_vector_type(2))) float v2f;
typedef __attribute__((ext_vector_type(8))) float v8f;
typedef __attribute__((__vector_size__(16))) int vi4;   // int4 vector for async-LDS builtin params

#if defined(__gfx1250__) && __has_builtin(__builtin_amdgcn_wmma_f32_16x16x4_f32)
#define HAVE_WMMA_F32 1
#else
#define HAVE_WMMA_F32 0
#endif

#if defined(__gfx1250__) && __has_builtin(__builtin_amdgcn_global_load_async_to_lds_b128)
#define HAVE_ASYNC_LDS 1
#else
#define HAVE_ASYNC_LDS 0
#endif

__global__ __launch_bounds__(256) void
gather_sort_mask_kernel(const float* __restrict__ hid,
                        const float* __restrict__ dis,
                        float* __restrict__ out) {
  extern __shared__ char smem[];
  float* hidS = (float*)smem;                      // [64][HP] staged hid[r]
  int*   selS = (int*)(smem + N_AGENT * HP * 4);   // [AG_PER_BLK][64]: src row or -1

  const int tid  = threadIdx.x;
  const int lane = tid & 31;
  const int w    = tid >> 5;       // wave 0..7
  const int lhi  = lane >> 4;      // half-wave 0/1
  const int llo  = lane & 15;

  const int blk    = blockIdx.x;
  const int r      = blk / SPLIT;
  const int agBase = (blk % SPLIT) * AG_PER_BLK;

  // ---------- Phase 1: stage hid[r] (64 x 256 f32 = 64 KB) into LDS ----------
  const float* hg = hid + (size_t)r * (N_AGENT * HIDDEN);
#if HAVE_ASYNC_LDS
  for (int i = tid; i < N_AGENT * (HIDDEN / 4); i += 256) {
    const int row = i >> 6, c4 = i & 63;
    __builtin_amdgcn_global_load_async_to_lds_b128(
        (vi4*)(hg + row * HIDDEN + c4 * 4),
        (vi4*)(hidS + row * HP + c4 * 4), 0, 0);
  }
#else
  for (int i = tid; i < N_AGENT * (HIDDEN / 4); i += 256) {
    const int row = i >> 6, c4 = i & 63;
    const float4 v = *(const float4*)(hg + row * HIDDEN + c4 * 4);
    *(float4*)(hidS + row * HP + c4 * 4) = v;
  }
#endif

  // ---------- Phase 2 (overlaps async stage): stable rank of 64 distances ----------
  // Wave w handles agents {2w, 2w+1}. Lane owns keys j=lane and j=lane+32.
  for (int al = 2 * w; al < 2 * w + 2; ++al) {
    const float* dr = dis + ((size_t)(r * N_AGENT) + agBase + al) * N_AGENT;
    const float d0 = dr[lane];
    const float d1 = dr[lane + 32];
    int r0 = 0, r1 = 0;
#pragma unroll 8
    for (int k = 0; k < 32; ++k) {
      const float ka = __shfl(d0, k, 32);   // key index k
      const float kb = __shfl(d1, k, 32);   // key index k+32
      // stable order: (value, index) lexicographic, matching jnp.argsort
      r0 += (ka < d0) || (ka == d0 && k < lane);   // vs j=lane
      r0 += (kb < d0);                              // (k+32) < lane impossible -> strict only
      r1 += (ka <= d1);                             // k < lane+32 always -> ties count
      r1 += (kb < d1) || (kb == d1 && k < lane);    // (k+32) < (lane+32) <=> k < lane
    }
    selS[al * 64 + r0] = ((d0 < CL_DIST) || (r0 < 2)) ? lane : -1;
    selS[al * 64 + r1] = ((d1 < CL_DIST) || (r1 < 2)) ? (lane + 32) : -1;
  }

#if HAVE_ASYNC_LDS
#if __has_builtin(__builtin_amdgcn_s_wait_asynccnt)
  __builtin_amdgcn_s_wait_asynccnt(0);
#else
  asm volatile("s_wait_asynccnt 0" ::: "memory");
#endif
#endif
  __syncthreads();

  // ---------- Phase 3: out_tile = P x hid[r] via V_WMMA_F32_16X16X4_F32 ----------
  // M = output rank a (4 tiles/agent), N = hidden h (16 tiles), K = source row j (16 steps of 4).
  // Per wave: 16 N-tiles x 8 (agent, M-tile) pairs; B K-slab (64x16) held in 32 VGPRs, reused.
  for (int nt = 0; nt < 16; ++nt) {
    const int h0 = nt * 16;
    v2f b[16];
#pragma unroll
    for (int ks = 0; ks < 16; ++ks) {
      // B 4x16 f32 layout: VGPR v, half lhi -> K = 4*ks + 2*lhi + v, N = llo
      b[ks].x = hidS[(4 * ks + 2 * lhi)     * HP + h0 + llo];
      b[ks].y = hidS[(4 * ks + 2 * lhi + 1) * HP + h0 + llo];
    }
    for (int p = 0; p < 8; ++p) {
      const int pairIdx = w + 8 * p;   // 0..63
      const int ag = pairIdx >> 2;     // 0..15
      const int mt = pairIdx & 3;      // 0..3
#if HAVE_WMMA_F32
      // A 16x4 f32 layout: lane -> M = llo (both halves); VGPR v, half lhi -> K = 2*lhi + v
      const int jsel = selS[ag * 64 + mt * 16 + llo];   // src row for output rank row M
      v8f c = {0.f, 0.f, 0.f, 0.f, 0.f, 0.f, 0.f, 0.f};
#pragma unroll
      for (int ks = 0; ks < 16; ++ks) {
        const int kb = 4 * ks + 2 * lhi;
        v2f a;
        a.x = (jsel == kb)     ? 1.0f : 0.0f;   // one-hot permutation row (exact in f32)
        a.y = (jsel == kb + 1) ? 1.0f : 0.0f;
        c = __builtin_amdgcn_wmma_f32_16x16x4_f32(false, a, false, b[ks],
                                                  (short)0, c, false, false);
      }
#else
      v8f c;
#pragma unroll
      for (int v = 0; v < 8; ++v) {
        const int js = selS[ag * 64 + mt * 16 + v + 8 * lhi];
        c[v] = (js >= 0) ? hidS[js * HP + h0 + llo] : 0.0f;
      }
#endif
      // C/D 16x16 f32 layout: VGPR v, half lhi -> row M = v + 8*lhi, col N = llo
      float* op = out +
                  (((size_t)(r * N_AGENT + agBase + ag)) * N_AGENT + mt * 16) * HIDDEN +
                  h0 + llo;
#pragma unroll
      for (int v = 0; v < 8; ++v) {
        op[(size_t)(v + 8 * lhi) * HIDDEN] = c[v];  // two contiguous 64B segments / store
      }
    }
  }
}

extern "C" void kernel_launch(void* const* d_in, const int* in_sizes, int n_in,
                              void* d_out, int out_size, void* d_ws, size_t ws_size,
                              hipStream_t stream) {
  (void)in_sizes; (void)n_in; (void)out_size; (void)d_ws; (void)ws_size;
  const float* hid = (const float*)d_in[0];
  const float* dis = (const float*)d_in[1];
  float* out = (float*)d_out;
  gather_sort_mask_kernel<<<dim3(N_ROLL * SPLIT), dim3(256), SMEM_BYTES, stream>>>(hid, dis, out);
}